// EntropyBottleneck_82343112999387
// MI455X (gfx1250) — compile-verified
//
#include <hip/hip_runtime.h>
#include <hip/hip_bf16.h>
#include <stdint.h>

// ---------------------------------------------------------------------------
// EntropyBottleneck likelihood kernel for MI455X (gfx1250).
//  - TDM (tensor_load_to_lds) double-buffered input staging (ASYNC path,
//    s_wait_tensorcnt), ds_read_b128 consumption.
//  - Per-lane FMA chains for the per-channel (1,3,3,3,3,1) MLP; v_tanh_f32.
//  - Affine fast path: when all tanh(t_i)==0 the chain is exactly affine;
//    fold to y = a*x + b per channel (uniform branch, fully general).
//  - Non-temporal b128 stores for the 603 MB of write-once output.
// ---------------------------------------------------------------------------

typedef float v4f __attribute__((ext_vector_type(4)));

#define EB_BLOCK        256
#define EB_STAGE_FLOATS 4096            // 16 KB per stage
#define EB_STAGES       4               // 16384 floats per block span
#define EB_STAGE_V4     (EB_STAGE_FLOATS / 4)
#define EB_ITERS        (EB_STAGE_V4 / EB_BLOCK)

#define LOG2E_F 1.4426950408889634f
#define LN2_F   0.6931471805599453f

// -------- device-side feature probes ---------------------------------------
#if defined(__HIP_DEVICE_COMPILE__) && \
    __has_builtin(__builtin_amdgcn_tensor_load_to_lds) && \
    __has_builtin(__builtin_amdgcn_s_wait_tensorcnt)
#define EB_DEV_TDM 1
#else
#define EB_DEV_TDM 0
#endif

__device__ __forceinline__ float fexp2(float x) { return __builtin_amdgcn_exp2f(x); }
__device__ __forceinline__ float frcp(float x)  { return __builtin_amdgcn_rcpf(x); }
__device__ __forceinline__ float flog2(float x) { return __builtin_amdgcn_logf(x); }

__device__ __forceinline__ float fast_tanh(float x) {
#if __has_builtin(__builtin_amdgcn_tanhf)
  return __builtin_amdgcn_tanhf(x);            // gfx1250 v_tanh_f32
#elif __has_builtin(__builtin_amdgcn_tanh_f32)
  return __builtin_amdgcn_tanh_f32(x);
#else
  float e = fexp2(x * (2.0f * LOG2E_F));       // tanh(x) = 1 - 2/(e^{2x}+1)
  return 1.0f - 2.0f * frcp(e + 1.0f);
#endif
}

__device__ __forceinline__ float fast_sigmoid(float x) {
  return frcp(1.0f + fexp2(-LOG2E_F * x));
}

__device__ __forceinline__ float softplus_f(float x) {
  // stable: max(x,0) + log1p(exp(-|x|))
  float e = fexp2(-LOG2E_F * fabsf(x));
  return fmaxf(x, 0.0f) + flog2(1.0f + e) * LN2_F;
}

// -------- per-channel transformed weights (all block-uniform -> s_loads) ----
struct CW {
  float w0[3], w1[9], w2[9], w3[9], w4[3];
  float bb0[3], bb1[3], bb2[3], bb3[3], bb4;
  float g0[3], g1[3], g2[3], g3[3];
  int hg0, hg1, hg2, hg3;
};

__device__ __forceinline__ void load_cw(CW& W, int c,
    const float* __restrict__ m0, const float* __restrict__ pb0, const float* __restrict__ t0,
    const float* __restrict__ m1, const float* __restrict__ pb1, const float* __restrict__ t1,
    const float* __restrict__ m2, const float* __restrict__ pb2, const float* __restrict__ t2,
    const float* __restrict__ m3, const float* __restrict__ pb3, const float* __restrict__ t3,
    const float* __restrict__ m4, const float* __restrict__ pb4)
{
#pragma unroll
  for (int j = 0; j < 3; ++j) {
    W.w0[j]  = softplus_f(m0[c * 3 + j]);
    W.w4[j]  = softplus_f(m4[c * 3 + j]);
    W.bb0[j] = pb0[c * 3 + j];
    W.bb1[j] = pb1[c * 3 + j];
    W.bb2[j] = pb2[c * 3 + j];
    W.bb3[j] = pb3[c * 3 + j];
    W.g0[j]  = fast_tanh(t0[c * 3 + j]);
    W.g1[j]  = fast_tanh(t1[c * 3 + j]);
    W.g2[j]  = fast_tanh(t2[c * 3 + j]);
    W.g3[j]  = fast_tanh(t3[c * 3 + j]);
  }
#pragma unroll
  for (int j = 0; j < 9; ++j) {
    W.w1[j] = softplus_f(m1[c * 9 + j]);
    W.w2[j] = softplus_f(m2[c * 9 + j]);
    W.w3[j] = softplus_f(m3[c * 9 + j]);
  }
  W.bb4 = pb4[c];
  W.hg0 = (W.g0[0] != 0.f) | (W.g0[1] != 0.f) | (W.g0[2] != 0.f);
  W.hg1 = (W.g1[0] != 0.f) | (W.g1[1] != 0.f) | (W.g1[2] != 0.f);
  W.hg2 = (W.g2[0] != 0.f) | (W.g2[1] != 0.f) | (W.g2[2] != 0.f);
  W.hg3 = (W.g3[0] != 0.f) | (W.g3[1] != 0.f) | (W.g3[2] != 0.f);
}

// When every gate is zero the chain is affine: fold to y = a*x + b.
__device__ __forceinline__ void fold_affine(const CW& W, float& a, float& b) {
  float v0 = W.w0[0], v1 = W.w0[1], v2 = W.w0[2];
  float u0 = W.bb0[0], u1 = W.bb0[1], u2 = W.bb0[2];
#define EB_FOLD(Wm, Bb)                                                    \
  {                                                                        \
    float nv0 = fmaf(Wm[0], v0, fmaf(Wm[1], v1, Wm[2] * v2));              \
    float nv1 = fmaf(Wm[3], v0, fmaf(Wm[4], v1, Wm[5] * v2));              \
    float nv2 = fmaf(Wm[6], v0, fmaf(Wm[7], v1, Wm[8] * v2));              \
    float nu0 = fmaf(Wm[0], u0, fmaf(Wm[1], u1, fmaf(Wm[2], u2, Bb[0]))); \
    float nu1 = fmaf(Wm[3], u0, fmaf(Wm[4], u1, fmaf(Wm[5], u2, Bb[1]))); \
    float nu2 = fmaf(Wm[6], u0, fmaf(Wm[7], u1, fmaf(Wm[8], u2, Bb[2]))); \
    v0 = nv0; v1 = nv1; v2 = nv2; u0 = nu0; u1 = nu1; u2 = nu2;            \
  }
  EB_FOLD(W.w1, W.bb1)
  EB_FOLD(W.w2, W.bb2)
  EB_FOLD(W.w3, W.bb3)
#undef EB_FOLD
  a = fmaf(W.w4[0], v0, fmaf(W.w4[1], v1, W.w4[2] * v2));
  b = fmaf(W.w4[0], u0, fmaf(W.w4[1], u1, fmaf(W.w4[2], u2, W.bb4)));
}

__device__ __forceinline__ float eval_chain(const CW& W, float x) {
  float y0 = fmaf(W.w0[0], x, W.bb0[0]);
  float y1 = fmaf(W.w0[1], x, W.bb0[1]);
  float y2 = fmaf(W.w0[2], x, W.bb0[2]);
  if (W.hg0) {
    y0 = fmaf(W.g0[0], fast_tanh(y0), y0);
    y1 = fmaf(W.g0[1], fast_tanh(y1), y1);
    y2 = fmaf(W.g0[2], fast_tanh(y2), y2);
  }
#define EB_LAYER(Wm, Bb, Gg, Hg)                                          \
  {                                                                       \
    float z0 = fmaf(Wm[0], y0, fmaf(Wm[1], y1, fmaf(Wm[2], y2, Bb[0]))); \
    float z1 = fmaf(Wm[3], y0, fmaf(Wm[4], y1, fmaf(Wm[5], y2, Bb[1]))); \
    float z2 = fmaf(Wm[6], y0, fmaf(Wm[7], y1, fmaf(Wm[8], y2, Bb[2]))); \
    if (Hg) {                                                             \
      z0 = fmaf(Gg[0], fast_tanh(z0), z0);                                \
      z1 = fmaf(Gg[1], fast_tanh(z1), z1);                                \
      z2 = fmaf(Gg[2], fast_tanh(z2), z2);                                \
    }                                                                     \
    y0 = z0; y1 = z1; y2 = z2;                                            \
  }
  EB_LAYER(W.w1, W.bb1, W.g1, W.hg1)
  EB_LAYER(W.w2, W.bb2, W.g2, W.hg2)
  EB_LAYER(W.w3, W.bb3, W.g3, W.hg3)
#undef EB_LAYER
  return fmaf(W.w4[0], y0, fmaf(W.w4[1], y1, fmaf(W.w4[2], y2, W.bb4)));
}

template <bool LIN>
__device__ __forceinline__ void eval_elem(const CW& W, float Aa, float bl, float bu,
                                          float x, float& lk, float& lo, float& up) {
  if (LIN) {
    lo = fmaf(Aa, x, bl);
    up = fmaf(Aa, x, bu);
  } else {
    lo = eval_chain(W, x - 0.5f);
    up = eval_chain(W, x + 0.5f);
  }
  lk = fast_sigmoid(up) - fast_sigmoid(lo);
}

// -------- Tensor Data Mover: 1-D tile load, D# built per ISA §8 -------------
#if EB_DEV_TDM
typedef unsigned int u32x4 __attribute__((ext_vector_type(4)));
typedef int i32x8 __attribute__((ext_vector_type(8)));
typedef int i32x4 __attribute__((ext_vector_type(4)));

__device__ __forceinline__ void tdm_load_1d(const float* gsrc,
                                            unsigned lds_byte_off,
                                            unsigned nelem) {
  unsigned long long ga = (unsigned long long)(uintptr_t)gsrc;
  u32x4 g0;
  g0.x = 1u;                                            // count=1, user load
  g0.y = lds_byte_off;                                  // lds_addr (bytes)
  g0.z = (unsigned)(ga & 0xffffffffu);                  // global_addr[31:0]
  g0.w = (unsigned)((ga >> 32) & 0x1ffffffu)            // global_addr[56:32]
       | (2u << 30);                                    // type=2 ("image")
  i32x8 g1;
  g1[0] = (int)(2u << 16);                              // data_size=4B, mask=0
  g1[1] = (int)((nelem & 0xffffu) << 16);               // tensor_dim0[15:0]
  g1[2] = (int)((nelem >> 16) | (1u << 16));            // dim0 hi | tensor_dim1=1
  g1[3] = (int)((nelem & 0xffffu) << 16);               // tile_dim0
  g1[4] = 1;                                            // tile_dim1=1, tile_dim2=0
  g1[5] = (int)nelem;                                   // tensor_dim0_stride lo
  g1[6] = 0;                                            // stride0 hi | stride1 lo
  g1[7] = 0;                                            // stride1 hi
  i32x4 g2; g2[0] = 1; g2[1] = 1; g2[2] = 0; g2[3] = 0; // dim2=1, dim3=1
  i32x4 g3; g3[0] = 0; g3[1] = (1 << 16); g3[2] = 0; g3[3] = 0; // dim4=1
#if __clang_major__ >= 23
  i32x8 g4 = {0, 0, 0, 0, 0, 0, 0, 0};
  __builtin_amdgcn_tensor_load_to_lds(g0, g1, g2, g3, g4, 0);
#else
  __builtin_amdgcn_tensor_load_to_lds(g0, g1, g2, g3, 0);
#endif
}
#endif  // EB_DEV_TDM

// -------- stage pipeline (templated on affine fast path) --------------------
template <bool LIN>
__device__ __forceinline__ void run_span(const CW& W, float Aa, float bl, float bu,
                                         const float* __restrict__ xrow,
                                         float* __restrict__ oLk,
                                         float* __restrict__ oLo,
                                         float* __restrict__ oUp,
                                         int base, int N
#if EB_DEV_TDM
                                         , float* smem
#endif
                                         ) {
#if EB_DEV_TDM
  if (threadIdx.x < 32 && base < N) {
    unsigned cnt = (unsigned)min(EB_STAGE_FLOATS, N - base);
    tdm_load_1d(xrow + base, 0u, cnt);                   // prologue stage 0
  }
#endif

  for (int s = 0; s < EB_STAGES; ++s) {
    const int cur = base + s * EB_STAGE_FLOATS;          // block-uniform
    if (cur >= N) break;

#if EB_DEV_TDM
    if (threadIdx.x < 32) {
      const int nxt = cur + EB_STAGE_FLOATS;
      if (s + 1 < EB_STAGES && nxt < N) {
        unsigned cnt = (unsigned)min(EB_STAGE_FLOATS, N - nxt);
        tdm_load_1d(xrow + nxt,
                    (unsigned)(((s + 1) & 1) * EB_STAGE_FLOATS * sizeof(float)),
                    cnt);
        __builtin_amdgcn_s_wait_tensorcnt(1);            // stage s done
      } else {
        __builtin_amdgcn_s_wait_tensorcnt(0);
      }
    }
    __syncthreads();
    const float* __restrict__ src = smem + (s & 1) * EB_STAGE_FLOATS;
#else
    const float* __restrict__ src = xrow + cur;
#endif

#pragma unroll
    for (int it = 0; it < EB_ITERS; ++it) {
      const int lv = it * EB_BLOCK + (int)threadIdx.x;   // float4 idx in stage
      const int g = cur + lv * 4;                        // global float idx
      if (g + 3 < N) {
#if EB_DEV_TDM
        v4f x = *((const v4f*)src + lv);                 // ds_read_b128
#else
        v4f x = __builtin_nontemporal_load((const v4f*)src + lv);
#endif
        v4f lk, lo, up;
#pragma unroll
        for (int k = 0; k < 4; ++k) {
          float l, u, p;
          eval_elem<LIN>(W, Aa, bl, bu, x[k], p, l, u);
          lk[k] = p; lo[k] = l; up[k] = u;
        }
        const int v = g >> 2;
        __builtin_nontemporal_store(lk, (v4f*)oLk + v);
        __builtin_nontemporal_store(lo, (v4f*)oLo + v);
        __builtin_nontemporal_store(up, (v4f*)oUp + v);
      } else if (g < N) {
        for (int k = 0; k < N - g; ++k) {
          float xx = src[lv * 4 + k];
          float l, u, p;
          eval_elem<LIN>(W, Aa, bl, bu, xx, p, l, u);
          oLk[g + k] = p; oLo[g + k] = l; oUp[g + k] = u;
        }
      }
    }
#if EB_DEV_TDM
    __syncthreads();                                     // buf reuse fence
#endif
  }
}

// -------- main fused kernel -------------------------------------------------
__global__ __launch_bounds__(EB_BLOCK) void eb_fused(
    const float* __restrict__ inp,
    const float* __restrict__ m0, const float* __restrict__ pb0, const float* __restrict__ t0,
    const float* __restrict__ m1, const float* __restrict__ pb1, const float* __restrict__ t1,
    const float* __restrict__ m2, const float* __restrict__ pb2, const float* __restrict__ t2,
    const float* __restrict__ m3, const float* __restrict__ pb3, const float* __restrict__ t3,
    const float* __restrict__ m4, const float* __restrict__ pb4,
    float* __restrict__ out, int N, int C)
{
  const int c = blockIdx.y;

  CW W;
  load_cw(W, c, m0, pb0, t0, m1, pb1, t1, m2, pb2, t2, m3, pb3, t3, m4, pb4);

  const size_t CN = (size_t)C * (size_t)N;
  const float* __restrict__ xrow = inp + (size_t)c * (size_t)N;
  float* __restrict__ oLk = out + (size_t)c * (size_t)N;   // likelihood
  float* __restrict__ oLo = oLk + CN;                      // lower
  float* __restrict__ oUp = oLo + CN;                      // upper

  const int per_block = EB_STAGE_FLOATS * EB_STAGES;
  const int base = blockIdx.x * per_block;

#if EB_DEV_TDM
  extern __shared__ float smem[];                          // 2 x 16 KB
#endif

  // Block-uniform branch (depends only on blockIdx.y's weights): the chain is
  // exactly affine when all gates are zero (the reference's initialization).
  const bool lin = !(W.hg0 | W.hg1 | W.hg2 | W.hg3);
  if (lin) {
    float Aa, Bb;
    fold_affine(W, Aa, Bb);
    const float bl = fmaf(-0.5f, Aa, Bb);                  // B - A/2
    const float bu = fmaf( 0.5f, Aa, Bb);                  // B + A/2
    run_span<true>(W, Aa, bl, bu, xrow, oLk, oLo, oUp, base, N
#if EB_DEV_TDM
                   , smem
#endif
                   );
  } else {
    run_span<false>(W, 0.f, 0.f, 0.f, xrow, oLk, oLo, oUp, base, N
#if EB_DEV_TDM
                    , smem
#endif
                    );
  }
}

// -------- host launcher -----------------------------------------------------
extern "C" void kernel_launch(void* const* d_in, const int* in_sizes, int n_in,
                              void* d_out, int out_size, void* d_ws, size_t ws_size,
                              hipStream_t stream) {
  (void)n_in; (void)out_size; (void)d_ws; (void)ws_size;
  const float* inp = (const float*)d_in[0];
  const float* m0  = (const float*)d_in[1];
  const float* pb0 = (const float*)d_in[2];
  const float* t0  = (const float*)d_in[3];
  const float* m1  = (const float*)d_in[4];
  const float* pb1 = (const float*)d_in[5];
  const float* t1  = (const float*)d_in[6];
  const float* m2  = (const float*)d_in[7];
  const float* pb2 = (const float*)d_in[8];
  const float* t2  = (const float*)d_in[9];
  const float* m3  = (const float*)d_in[10];
  const float* pb3 = (const float*)d_in[11];
  const float* t3  = (const float*)d_in[12];
  const float* m4  = (const float*)d_in[13];
  const float* pb4 = (const float*)d_in[14];
  float* out = (float*)d_out;

  const int C = in_sizes[14];                 // b4 has C elements
  const int N = in_sizes[0] / C;

  const int per_block = EB_STAGE_FLOATS * EB_STAGES;
  dim3 grid((unsigned)((N + per_block - 1) / per_block), (unsigned)C, 1);
  size_t shmem = 2u * EB_STAGE_FLOATS * sizeof(float);    // TDM double buffer

  eb_fused<<<grid, dim3(EB_BLOCK), shmem, stream>>>(
      inp, m0, pb0, t0, m1, pb1, t1, m2, pb2, t2, m3, pb3, t3, m4, pb4,
      out, N, C);
}